// QuanvolutionLayer_47837345742891
// MI455X (gfx1250) — compile-verified
//
#include <hip/hip_runtime.h>
#include <hip/hip_bf16.h>
#include <math.h>

// Quanvolution layer for MI455X (gfx1250, wave32).
// y = U(16x16) * Psi(16x16 patches) via 4x chained V_WMMA_F32_16X16X4_F32.

typedef __attribute__((ext_vector_type(2))) float v2f;
typedef __attribute__((ext_vector_type(8))) float v8f;

#define KSZ  2
#define NQ   4
#define QDIM 16
#define IH   224
#define IW   224
#define LX   223
#define LY   223

// CNOT ring permutation on basis index k (qubit 0 = MSB):
// CNOT(0,1) CNOT(1,2) CNOT(2,3) CNOT(3,0)
__device__ __forceinline__ int perm_cnot(int k) {
    int b0 = (k >> 3) & 1;  // qubit 0
    int b1 = (k >> 2) & 1;  // qubit 1
    int b2 = (k >> 1) & 1;  // qubit 2
    int b3 =  k       & 1;  // qubit 3
    b1 ^= b0; b2 ^= b1; b3 ^= b2; b0 ^= b3;
    return (b0 << 3) | (b1 << 2) | (b2 << 1) | b3;
}

__global__ __launch_bounds__(256) void quanv_wmma_kernel(
    const float* __restrict__ in,    // (64,1,224,224)
    const float* __restrict__ wgt,   // (1,4)
    float* __restrict__ out,         // (64,4,223,223)
    int nwaves)
{
    const int lane = threadIdx.x & 31;
    const int wid  = (int)((blockIdx.x * blockDim.x + threadIdx.x) >> 5);
    if (wid >= nwaves) return;               // wave-uniform (exact grid anyway)

    const int batch = wid / LX;
    const int x     = wid % LX;              // patch row
    const int m     = lane & 15;             // matrix row / patch column in group
    const int hi    = lane >> 4;             // half selector

    // ---- Build this lane's 8 entries of U = CNOT_RING @ kron(RY(w_q)) ----
    float cw[NQ], sw[NQ];
#pragma unroll
    for (int q = 0; q < NQ; ++q) {
        float wq = wgt[q];                   // scalar (uniform) loads
        __sincosf(0.5f * wq, &sw[q], &cw[q]);
    }
    // U[i][j] = Ukron[perm^{-1}(i)][j]; find src = perm^{-1}(m)
    int src = 0;
#pragma unroll
    for (int t = 0; t < QDIM; ++t)
        if (perm_cnot(t) == m) src = t;

    float Ureg[8];                           // A-matrix VGPRs, 4 K-blocks x 2
#pragma unroll
    for (int k = 0; k < 4; ++k) {
#pragma unroll
        for (int j = 0; j < 2; ++j) {
            int col = 4 * k + 2 * hi + j;    // column index bits (b3 b2 b1 b0)
            float v = 1.0f;
#pragma unroll
            for (int q = 0; q < NQ; ++q) {
                int sq = (src >> (3 - q)) & 1;   // row bit of RY(w_q)
                int cq = (col >> (3 - q)) & 1;   // col bit
                // RY = [[c, -s],[s, c]]
                float f = (sq == cq) ? cw[q] : ((sq == 0) ? -sw[q] : sw[q]);
                v *= f;
            }
            Ureg[2 * k + j] = v;
        }
    }

    const float* inb  = in  + (size_t)batch * IH * IW;
    float*       outb = out + (size_t)batch * NQ * LX * LY;
    const float  HALF_PI = 1.57079632679489662f;

    // ---- 14 groups of 16 patches along this row ----
    for (int y0 = 0; y0 < LY; y0 += 16) {
        int py  = y0 + m;
        int pyc = (py < LY) ? py : (LY - 1);     // clamp loads for tail lane

        const float* r0 = inb + (size_t)x * IW + pyc;
        const float* r1 = r0 + IW;
        float p00 = r0[0], p01 = r0[1];          // qubits 0,1 (row x)
        float p10 = r1[0], p11 = r1[1];          // qubits 2,3 (row x+1)

        // RY(pi*x)|0> amplitudes: (cos(pi*x/2), sin(pi*x/2))
        float c0, s0, c1, s1, c2, s2, c3, s3;
        __sincosf(p00 * HALF_PI, &s0, &c0);
        __sincosf(p01 * HALF_PI, &s1, &c1);
        __sincosf(p10 * HALF_PI, &s2, &c2);
        __sincosf(p11 * HALF_PI, &s3, &c3);

        float t2 = hi ? s2 : c2;                 // qubit-2 factor (b1 = hi)
        float mk0 = (c0 * c1) * t2;              // k = (b3 b2) block factors
        float mk1 = (c0 * s1) * t2;
        float mk2 = (s0 * c1) * t2;
        float mk3 = (s0 * s1) * t2;
        float mk[4] = { mk0, mk1, mk2, mk3 };

        // y = U * psi : four chained f32 WMMAs (K = 4 each)
        v8f acc = { 0.f, 0.f, 0.f, 0.f, 0.f, 0.f, 0.f, 0.f };
#pragma unroll
        for (int k = 0; k < 4; ++k) {
            v2f a = { Ureg[2 * k], Ureg[2 * k + 1] };
            v2f b = { mk[k] * c3, mk[k] * s3 };  // b0 = 0 -> c3, b0 = 1 -> s3
            acc = __builtin_amdgcn_wmma_f32_16x16x4_f32(
                false, a, false, b, (short)0, acc, false, false);
        }

        // Born rule + Z-expectations: partial signed sums over this half's rows
        float pz0 = 0.f, pz1 = 0.f, pz2 = 0.f, pz3 = 0.f;
#pragma unroll
        for (int r = 0; r < 8; ++r) {
            float pr = acc[r] * acc[r];
            int g = hi * 8 + r;                  // global row (basis index)
            pz0 += ((g >> 3) & 1) ? -pr : pr;    // qubit 0 sign = bit 3
            pz1 += ((g >> 2) & 1) ? -pr : pr;
            pz2 += ((g >> 1) & 1) ? -pr : pr;
            pz3 += ( g       & 1) ? -pr : pr;
        }
        // combine lane n (rows 0-7) with lane n+16 (rows 8-15)
        pz0 += __shfl_xor(pz0, 16, 32);
        pz1 += __shfl_xor(pz1, 16, 32);
        pz2 += __shfl_xor(pz2, 16, 32);
        pz3 += __shfl_xor(pz3, 16, 32);

        if (py < LY) {                           // masked stores after WMMA
            size_t plane = (size_t)LX * LY;
            size_t o = (size_t)x * LY + py;
            if (hi == 0) {
                outb[o]             = pz0;       // q = 0
                outb[o + plane]     = pz1;       // q = 1
            } else {
                outb[o + 2 * plane] = pz2;       // q = 2
                outb[o + 3 * plane] = pz3;       // q = 3
            }
        }
    }
}

extern "C" void kernel_launch(void* const* d_in, const int* in_sizes, int n_in,
                              void* d_out, int out_size, void* d_ws, size_t ws_size,
                              hipStream_t stream) {
    (void)n_in; (void)d_ws; (void)ws_size; (void)out_size;
    const float* in  = (const float*)d_in[0];   // (64,1,224,224) fp32
    const float* wgt = (const float*)d_in[1];   // (1,4) fp32
    float*       out = (float*)d_out;           // (64,4,223,223) fp32

    int B = in_sizes[0] / (IH * IW);            // 64
    int nwaves = B * LX;                        // 14272 -> exactly 1784 blocks
    int threads = 256;                          // 8 waves per block
    int blocks = (nwaves * 32 + threads - 1) / threads;
    quanv_wmma_kernel<<<blocks, threads, 0, stream>>>(in, wgt, out, nwaves);
}